// LagrangeKANNmaninner_11055245820074
// MI455X (gfx1250) — compile-verified
//
#include <hip/hip_runtime.h>

#define N_WIDTH 64
#define N_ORDER 4
#define N_ELEMENTS 250
#define N_NODES (N_ELEMENTS * N_ORDER + 1)   // 1001
#define N_COLL 200
#define NDIM 2
#define M_ORD (N_ORDER + 1)                  // 5
#define ROW_LEN (N_NODES * NDIM)             // 2002 = 500*4 + 2
#define ROW_MAIN 2000                        // multiple-of-4 part
#define BUF_ELEMS ((size_t)N_COLL * N_WIDTH * N_NODES * NDIM)  // 25,625,600
#define OUT_HDR (3 * N_WIDTH)                // 192 floats

typedef __attribute__((ext_vector_type(2))) float v2f;
typedef __attribute__((ext_vector_type(4))) float v4f;   // native clang vector (B128)
typedef __attribute__((ext_vector_type(8))) float v8f;

// ---------------------------------------------------------------------------
// Kernel 1: stream-copy the three big buffers into d_out (non-temporal: the
// 307MB stream must not evict reused data from the 192MB L2).
// ---------------------------------------------------------------------------
__global__ void copy_bufs_kernel(const v4f* __restrict__ s0,
                                 const v4f* __restrict__ s1,
                                 const v4f* __restrict__ s2,
                                 v4f* __restrict__ dst_base) {
  const int which = blockIdx.y;
  const v4f* src = (which == 0) ? s0 : ((which == 1) ? s1 : s2);
  const size_t n4 = BUF_ELEMS / 4;
  v4f* dst = dst_base + (size_t)which * n4;
  size_t i = (size_t)blockIdx.x * blockDim.x + threadIdx.x;
  if (i < n4) {
    v4f v = __builtin_nontemporal_load(&src[i]);
    __builtin_nontemporal_store(v, &dst[i]);
  }
}

// ---------------------------------------------------------------------------
// Lagrange basis (order 4, nodes linspace(-1,1,5)), reproducing the reference
// exactly, including the dL masking to p == M-1 only.
// ---------------------------------------------------------------------------
__device__ __forceinline__ void lag_basis(float xt, float L[5], float dL[5],
                                          float ddL[5]) {
  const float nodes[5] = {-1.0f, -0.5f, 0.0f, 0.5f, 1.0f};
  float inv[5][5], r[5][5];
#pragma unroll
  for (int i = 0; i < 5; ++i) {
#pragma unroll
    for (int j = 0; j < 5; ++j) {
      inv[i][j] = (i == j) ? 0.0f : 1.0f / (nodes[i] - nodes[j]);
      r[i][j] = (xt - nodes[j]) * inv[i][j];
    }
  }
#pragma unroll
  for (int i = 0; i < 5; ++i) {
    float prod = 1.0f;
#pragma unroll
    for (int j = 0; j < 5; ++j)
      if (j != i) prod *= r[i][j];
    L[i] = prod;

    float s1 = 0.0f;
#pragma unroll
    for (int k = 0; k < 5; ++k) {
      if (k == i) continue;
      float p = 1.0f;
#pragma unroll
      for (int j = 0; j < 5; ++j)
        if (j != i && j != k) p *= r[i][j];
      s1 += inv[i][k] * p;
    }
    dL[i] = (i == 4) ? s1 : 0.0f;  // reference masks dL to index M-1 only

    float s2 = 0.0f;
#pragma unroll
    for (int k = 0; k < 5; ++k) {
      if (k == i) continue;
      float ks = 0.0f;
#pragma unroll
      for (int l = 0; l < 5; ++l) {
        if (l == i || l == k) continue;
        float p = 1.0f;
#pragma unroll
        for (int j = 0; j < 5; ++j)
          if (j != i && j != k && j != l) p *= r[i][j];
        ks += inv[i][l] * p;
      }
      s2 += inv[i][k] * ks;
    }
    ddL[i] = s2;
  }
}

// ---------------------------------------------------------------------------
// Kernel 2: scatter. One thread per width-row w (all rows get identical
// values). Loops follow the reference order d -> p -> dq so the JAX
// .at[].set overwrite semantics (last write wins) are reproduced.
// ---------------------------------------------------------------------------
__global__ void scatter_kernel(const float* __restrict__ x,
                               const int* __restrict__ flag,
                               const int* __restrict__ sample,
                               float* __restrict__ out) {
  if (*flag != 0) return;
  const int w = threadIdx.x;  // 0..63
  const int smp = *sample;

  int nl[NDIM];
  float xt[NDIM];
#pragma unroll
  for (int d = 0; d < NDIM; ++d) {
    float xs = (float)(N_NODES - 1) * (x[d] - 0.0f) / 1.0f;
    float ide = floorf(xs / (float)N_ORDER);
    ide = fminf(fmaxf(ide, 0.0f), (float)(N_ELEMENTS - 1));
    int nlv = (int)(ide * (float)N_ORDER);
    float nlf = (float)nlv;
    float nrf = nlf + (float)N_ORDER;
    nl[d] = nlv;
    xt[d] = (xs - 0.5f * (nlf + nrf)) / (0.5f * (nrf - nlf));
  }

  const float dx = 0.002f;  // 0.5*N_ORDER*(X_MAX-X_MIN)/(N_NODES-1)
  float* phi = out + OUT_HDR;
  float* dphi = phi + BUF_ELEMS;
  float* ddphi = dphi + BUF_ELEMS;
  const size_t base = (((size_t)smp * N_WIDTH + (size_t)w) * N_NODES) * NDIM;

  for (int d = 0; d < NDIM; ++d) {
    float L[5], dL[5], ddL[5];
    lag_basis(xt[d], L, dL, ddL);
    for (int p = 0; p < M_ORD; ++p) {
      float v = L[p];
      float dv = dL[p] / dx;
      float ddv = ddL[p] / (dx * dx);
      for (int dq = 0; dq < NDIM; ++dq) {
        int pos = nl[dq] + p;
        size_t idx = base + (size_t)pos * NDIM + (size_t)d;
        phi[idx] = v;
        dphi[idx] = dv;
        ddphi[idx] = ddv;
      }
    }
  }
}

// ---------------------------------------------------------------------------
// Kernel 3: t[k] = sum_j W[k,j] * Buf[sample,k,j] for the three buffers,
// computed as the diagonal of W x Phi^T via V_WMMA_F32_16X16X4_F32 (full f32).
// One wave per (buffer, 16-row k tile): 12 waves total.
// A 16x4 f32 layout: lane L holds row M=L&15; VGPR{0,1} = K = 2*(L>>4)+{0,1}.
// C/D 16x16 f32 layout: lane L, VGPR v -> row M = v + 8*(L>>4), col N = L&15.
// Main loop (j < 2000) is fully unguarded: b64 loads, EXEC all-ones.
// Tail (j = 2000,2001): both halves load the same valid b64; the upper
// lane-half masks its contribution with *0.0f (branchless, no exec churn).
// ---------------------------------------------------------------------------
__global__ void einsum_wmma_kernel(const float* __restrict__ weight,
                                   const int* __restrict__ sample,
                                   float* __restrict__ out) {
  const int wid = blockIdx.x;        // 0..11
  const int b = wid >> 2;            // buffer 0..2
  const int k0 = (wid & 3) * 16;     // k-tile base
  const int lane = threadIdx.x;      // 0..31 (wave32)
  const int half = lane >> 4;
  const int n = lane & 15;

  const float* src = out + OUT_HDR + (size_t)b * BUF_ELEMS +
                     (size_t)(*sample) * N_WIDTH * ROW_LEN;
  const float* wrow = weight + (size_t)(k0 + n) * ROW_LEN;  // A row (M = n)
  const float* prow = src + (size_t)(k0 + n) * ROW_LEN;     // B col (N = n)
  const int koff = 2 * half;  // this lane's K offset within a 4-chunk

  v8f c = {};
  for (int j0 = 0; j0 < ROW_MAIN; j0 += 4) {
    v2f a  = *(const v2f*)(wrow + j0 + koff);
    v2f bf = *(const v2f*)(prow + j0 + koff);
    // D = A(16x4) * B(4x16) + C, all f32
    c = __builtin_amdgcn_wmma_f32_16x16x4_f32(
        /*neg_a=*/false, a, /*neg_b=*/false, bf,
        /*c_mod=*/(short)0, c, /*reuse_a=*/false, /*reuse_b=*/false);
  }
  {  // tail: K-chunk [2000..2003], only 2000/2001 exist
    const float m = (half == 0) ? 1.0f : 0.0f;
    v2f la = *(const v2f*)(wrow + ROW_MAIN);
    v2f lb = *(const v2f*)(prow + ROW_MAIN);
    v2f a, bf;
    a.x = la.x * m;
    a.y = la.y * m;
    bf.x = lb.x * m;
    bf.y = lb.y * m;
    c = __builtin_amdgcn_wmma_f32_16x16x4_f32(
        false, a, false, bf, (short)0, c, false, false);
  }

  // Extract diagonal element M == N == n.
  // half==0: rows 0..7 live in c[n] for n<8.  half==1: rows 8..15 in c[n-8].
  const bool writer = (half == 0) ? (n < 8) : (n >= 8);
  const int v = n - 8 * half;  // 0..7 for writer lanes
  float dval = 0.0f;
#pragma unroll
  for (int i = 0; i < 8; ++i) dval = (v == i) ? c[i] : dval;
  if (writer) out[(size_t)b * N_WIDTH + (size_t)(k0 + n)] = dval;
}

// ---------------------------------------------------------------------------
extern "C" void kernel_launch(void* const* d_in, const int* in_sizes, int n_in,
                              void* d_out, int out_size, void* d_ws,
                              size_t ws_size, hipStream_t stream) {
  (void)in_sizes; (void)n_in; (void)out_size; (void)d_ws; (void)ws_size;

  const float* x = (const float*)d_in[0];
  const int* flag = (const int*)d_in[1];
  const int* sample = (const int*)d_in[2];
  const float* weight = (const float*)d_in[3];
  const v4f* phi_in = (const v4f*)d_in[4];
  const v4f* dphi_in = (const v4f*)d_in[5];
  const v4f* ddphi_in = (const v4f*)d_in[6];
  float* out = (float*)d_out;

  // 1) Stream-copy the three returned buffers (bandwidth-bound: ~615 MB).
  const size_t n4 = BUF_ELEMS / 4;  // 6,406,400 v4f per buffer
  dim3 cgrid((unsigned)((n4 + 255) / 256), 3, 1);
  copy_bufs_kernel<<<cgrid, 256, 0, stream>>>(phi_in, dphi_in, ddphi_in,
                                              (v4f*)(out + OUT_HDR));

  // 2) Scatter the Lagrange basis values into the sample slice.
  scatter_kernel<<<1, 64, 0, stream>>>(x, flag, sample, out);

  // 3) Per-row dot products via f32 WMMA (diag of W x Phi^T), 12 waves.
  einsum_wmma_kernel<<<12, 32, 0, stream>>>(weight, sample, out);
}